// GraphConvolution_4045859193004
// MI455X (gfx1250) — compile-verified
//
#include <hip/hip_runtime.h>
#include <hip/hip_bf16.h>

typedef __attribute__((ext_vector_type(2))) float v2f;
typedef __attribute__((ext_vector_type(8))) float v8f;

// Problem constants (fixed by the reference)
constexpr int Lc = 2;
constexpr int Fc = 128;
constexpr int Oc = 128;
constexpr int Nc = 1024;
constexpr int Bc_ = 8;
constexpr int Dc = 3;
constexpr int O3 = 3 * Oc;

// LDS row pitch (dwords): multiple of 4 (16B-aligned rows for async b128) and
// 36*16 = 9*64 -> 16 consecutive rows hit 16 distinct banks (conflict-free).
constexpr int PITCH = 36;

union F2 { float2 f; v2f v; };

__device__ __forceinline__ void async_b128(const float* g, const float* l) {
    unsigned laddr = (unsigned)(unsigned long long)(uintptr_t)l;
    unsigned long long gaddr = (unsigned long long)(uintptr_t)g;
    asm volatile("global_load_async_to_lds_b128 %0, %1, off"
                 :: "v"(laddr), "v"(gaddr) : "memory");
}
__device__ __forceinline__ void async_b32(const float* g, const float* l) {
    unsigned laddr = (unsigned)(unsigned long long)(uintptr_t)l;
    unsigned long long gaddr = (unsigned long long)(uintptr_t)g;
    asm volatile("global_load_async_to_lds_b32 %0, %1, off"
                 :: "v"(laddr), "v"(gaddr) : "memory");
}
__device__ __forceinline__ void wait_async0() {
    asm volatile("s_wait_asynccnt 0x0" ::: "memory");
}

// ---------------------------------------------------------------------------
// Generic f32 WMMA GEMM:  C[map(m), n] = sum_k A[m,k] * B[k,n] (+bias[n]) (relu?)
//   - 64x64 tile per 256-thread workgroup (8 wave32s); wave w: rows w&3,
//     32-col slab w>>2 (two 16x16 accumulators)
//   - K chunks of 32, double-buffered in LDS via global_load_async_to_lds_*
//     (ASYNCcnt pipeline), 16 x v_wmma_f32_16x16x4_f32 per chunk per wave
//   - LDS layouts: A [row][k] pitch 36, B [col][k] pitch 36 -> every fragment
//     is one 8B-aligned ds_load_b64, conflict-free
//   - row->output map: idx = (m/rb)*s2 + (m%rb)*s1 + n ; wave row-base is a
//     multiple of rb (rb in {1,8}) so it collapses to qbase + r*rowStep + n
// ---------------------------------------------------------------------------
__global__ __launch_bounds__(256) void wmma_gemm_f32(
    const float* __restrict__ A, int lda, long long aBatch,
    const float* __restrict__ Bm, int ldb, long long bBatch, int transB,
    const float* __restrict__ bias, int biasBatch, int doRelu,
    float* __restrict__ C, long long cBatch, int rb, long long s1, long long s2,
    int M, int Nn, int Ktot)
{
    __shared__ float lsA[2][64 * PITCH];
    __shared__ float lsB[2][64 * PITCH];

    const int tid   = threadIdx.x;
    const int wave  = tid >> 5;
    const int lane  = tid & 31;
    const int half  = lane >> 4;   // 0: lanes 0-15, 1: lanes 16-31
    const int l16   = lane & 15;
    const int waveM = wave & 3;    // 16-row slab
    const int waveN2 = wave >> 2;  // 32-col slab

    const int m0 = blockIdx.y * 64;
    const int n0 = blockIdx.x * 64;
    const long long z = blockIdx.z;

    const float* Ab = A  + z * aBatch;
    const float* Bb = Bm + z * bBatch;

    // stage one K-chunk (32) into LDS buffers via async mem->LDS copies
    auto stage = [&](float* dA, float* dB, int k0) {
        #pragma unroll
        for (int p = 0; p < 2; ++p) {
            const int lin = p * 256 + tid;        // 0..511 x b128
            const int i = lin >> 3, kq = lin & 7;
            async_b128(Ab + (long long)(m0 + i) * lda + k0 + kq * 4,
                       dA + i * PITCH + kq * 4);
        }
        if (transB) {   // B(k,n) = Bg[n*ldb + k]: k-contiguous rows -> b128
            #pragma unroll
            for (int p = 0; p < 2; ++p) {
                const int lin = p * 256 + tid;
                const int n = lin >> 3, kq = lin & 7;
                async_b128(Bb + (long long)(n0 + n) * ldb + k0 + kq * 4,
                           dB + n * PITCH + kq * 4);
            }
        } else {        // B row-major: per-lane b32 does the transpose
            #pragma unroll
            for (int p = 0; p < 8; ++p) {
                const int lin = p * 256 + tid;    // 0..2047
                const int k = lin >> 6, n = lin & 63;   // coalesced in n
                async_b32(Bb + (long long)(k0 + k) * ldb + n0 + n,
                          dB + n * PITCH + k);
            }
        }
    };

    v8f acc[2];
    acc[0] = (v8f){0.f,0.f,0.f,0.f,0.f,0.f,0.f,0.f};
    acc[1] = (v8f){0.f,0.f,0.f,0.f,0.f,0.f,0.f,0.f};

    const int nChunks = Ktot >> 5;
    stage(lsA[0], lsB[0], 0);

    for (int c = 0; c < nChunks; ++c) {
        const float* curA = lsA[c & 1];
        const float* curB = lsB[c & 1];

        wait_async0();          // my async copies for chunk c are in LDS
        __syncthreads();        // everyone's are
        if (c + 1 < nChunks)    // prefetch next chunk into the other buffer
            stage(lsA[(c + 1) & 1], lsB[(c + 1) & 1], (c + 1) * 32);

        const float* apRow = curA + (waveM * 16 + l16) * PITCH + half * 2;
        const float* bpRow0 = curB + (waveN2 * 32 + l16) * PITCH + half * 2;
        const float* bpRow1 = bpRow0 + 16 * PITCH;

        #pragma unroll
        for (int kk = 0; kk < 8; ++kk) {
            // A frag: lane=M, VGPR j -> K = kk*4 + 2*half + j  (ISA 16x4 layout)
            F2 a;  a.f  = *(const float2*)(apRow  + kk * 4);
            F2 b0; b0.f = *(const float2*)(bpRow0 + kk * 4);
            F2 b1; b1.f = *(const float2*)(bpRow1 + kk * 4);
            acc[0] = __builtin_amdgcn_wmma_f32_16x16x4_f32(false, a.v, false, b0.v,
                                                           (short)0, acc[0], false, false);
            acc[1] = __builtin_amdgcn_wmma_f32_16x16x4_f32(false, a.v, false, b1.v,
                                                           (short)0, acc[1], false, false);
        }
        __syncthreads();        // all waves done with chunk c before restage
    }

    // ---- epilogue: bias, relu, mapped scatter store ----
    float* Cb = C + z * cBatch;
    const int hasBias = (bias != nullptr);
    const float* biasB = hasBias ? (bias + (long long)z * biasBatch) : nullptr;

    const int mBase = m0 + waveM * 16 + half * 8;         // multiple of rb
    const long long qbase = (long long)(mBase / rb) * s2;
    const long long rowStep = (rb == 1) ? s2 : s1;        // valid: 8 rows < rb*... rb in {1,8}

    #pragma unroll
    for (int t = 0; t < 2; ++t) {
        const int col = n0 + waveN2 * 32 + t * 16 + l16;
        const float bv = hasBias ? biasB[col] : 0.0f;
        #pragma unroll
        for (int r = 0; r < 8; ++r) {                     // D layout: VGPR r -> M = mBase + r
            float v = acc[t][r] + bv;
            if (doRelu) v = fmaxf(v, 0.0f);
            Cb[qbase + (long long)r * rowStep + col] = v;
        }
    }
}

// ---------------------------------------------------------------------------
// Fused GRU gates (per direction, hidden = x broadcast), sum over d, relu,
// highway gate combine.  One thread per (n,b,o) element.
// ---------------------------------------------------------------------------
__global__ __launch_bounds__(256) void gru_highway_fuse(
    const float* __restrict__ gi,   // [D,B,N,3O]
    const float* __restrict__ gh,   // [(n*B+b), 3O]
    const float* __restrict__ hw,   // [(n*B+b), O]  (relu(x@Wh+Bh) already)
    const float* __restrict__ x,    // [N,B,O]
    float* __restrict__ xout)       // [N,B,O]
{
    const int idx = blockIdx.x * blockDim.x + threadIdx.x;  // (n*B+b)*O + o
    if (idx >= Nc * Bc_ * Oc) return;
    const int o  = idx % Oc;
    const int nb = idx / Oc;
    const int b  = nb % Bc_;
    const int n  = nb / Bc_;

    const float xv  = x[idx];
    const float h_r = gh[(long long)nb * O3 + o];
    const float h_z = gh[(long long)nb * O3 + Oc + o];
    const float h_n = gh[(long long)nb * O3 + 2 * Oc + o];

    float accv = 0.0f;
    #pragma unroll
    for (int d = 0; d < Dc; ++d) {
        const long long base = (((long long)d * Bc_ + b) * Nc + n) * O3;
        const float i_r = gi[base + o];
        const float i_z = gi[base + Oc + o];
        const float i_n = gi[base + 2 * Oc + o];
        const float r  = 1.0f / (1.0f + __expf(-(i_r + h_r)));
        const float zz = 1.0f / (1.0f + __expf(-(i_z + h_z)));
        const float nn = tanhf(i_n + r * h_n);
        accv += (1.0f - zz) * nn + zz * xv;
    }
    const float outv = fmaxf(accv, 0.0f);
    const float t = hw[idx];
    xout[idx] = outv * t + xv * (1.0f - t);
}

extern "C" void kernel_launch(void* const* d_in, const int* in_sizes, int n_in,
                              void* d_out, int out_size, void* d_ws, size_t ws_size,
                              hipStream_t stream) {
    (void)in_sizes; (void)n_in; (void)out_size; (void)ws_size;

    const float* xin  = (const float*)d_in[0];  // [N,B,F]
    const float* adj  = (const float*)d_in[1];  // [D,B,N,N]
    const float* W    = (const float*)d_in[2];  // [L,D,F,O]
    const float* Bcb  = (const float*)d_in[3];  // [L,D,O]
    const float* Wh   = (const float*)d_in[4];  // [L,F,O]
    const float* Bh   = (const float*)d_in[5];  // [L,O]
    const float* Wih  = (const float*)d_in[6];  // [L,3O,O]
    const float* Whh  = (const float*)d_in[7];  // [L,3O,O]
    const float* bih  = (const float*)d_in[8];  // [L,3O]
    const float* bhh  = (const float*)d_in[9];  // [L,3O]

    float* ws      = (float*)d_ws;
    float* support = ws;                                      // [D,B,N,O]
    float* agg     = support + (size_t)Dc * Bc_ * Nc * Oc;    // [D,B,N,O]
    float* gh      = agg     + (size_t)Dc * Bc_ * Nc * Oc;    // [N*B, 3O]
    float* gi      = gh      + (size_t)Nc * Bc_ * O3;         // [D*B*N, 3O]
    float* hw      = gi      + (size_t)Dc * Bc_ * Nc * O3;    // [N*B, O]
    float* xbuf    = hw      + (size_t)Nc * Bc_ * Oc;         // [N,B,O]

    for (int l = 0; l < Lc; ++l) {
        const float* xsrc = (l == 0) ? xin : xbuf;
        float* xdst = (l == Lc - 1) ? (float*)d_out : xbuf;

        // K1: support[d,b,n,o] = x[n,b,:] @ W[l,d] + Bc[l,d]   (rows m=(n,b))
        {
            dim3 g(Oc / 64, (Nc * Bc_) / 64, Dc);
            wmma_gemm_f32<<<g, 256, 0, stream>>>(
                xsrc, Fc, 0LL,
                W + (long long)l * Dc * Fc * Oc, Oc, (long long)Fc * Oc, 0,
                Bcb + (long long)l * Dc * Oc, Oc, 0,
                support, (long long)Bc_ * Nc * Oc,
                Bc_, (long long)Nc * Oc, (long long)Oc,
                Nc * Bc_, Oc, Fc);
        }
        // K2: agg[d,b] = adj[d,b] @ support[d,b]   (the HBM-bound one)
        {
            dim3 g(Oc / 64, Nc / 64, Dc * Bc_);
            wmma_gemm_f32<<<g, 256, 0, stream>>>(
                adj, Nc, (long long)Nc * Nc,
                support, Oc, (long long)Nc * Oc, 0,
                nullptr, 0, 0,
                agg, (long long)Nc * Oc,
                1, 0LL, (long long)Oc,
                Nc, Oc, Nc);
        }
        // K3: gh[(n,b),:] = x @ Whh[l]^T + bhh[l]   (shared across d)
        {
            dim3 g(O3 / 64, (Nc * Bc_) / 64, 1);
            wmma_gemm_f32<<<g, 256, 0, stream>>>(
                xsrc, Fc, 0LL,
                Whh + (long long)l * O3 * Oc, Oc, 0LL, 1,
                bhh + (long long)l * O3, 0, 0,
                gh, 0LL, 1, 0LL, (long long)O3,
                Nc * Bc_, O3, Oc);
        }
        // K4: gi[(d,b,n),:] = agg @ Wih[l]^T + bih[l]
        {
            dim3 g(O3 / 64, (Dc * Bc_ * Nc) / 64, 1);
            wmma_gemm_f32<<<g, 256, 0, stream>>>(
                agg, Oc, 0LL,
                Wih + (long long)l * O3 * Oc, Oc, 0LL, 1,
                bih + (long long)l * O3, 0, 0,
                gi, 0LL, 1, 0LL, (long long)O3,
                Dc * Bc_ * Nc, O3, Oc);
        }
        // K5: hw[(n,b),:] = relu(x @ Wh[l] + Bh[l])
        {
            dim3 g(Oc / 64, (Nc * Bc_) / 64, 1);
            wmma_gemm_f32<<<g, 256, 0, stream>>>(
                xsrc, Fc, 0LL,
                Wh + (long long)l * Fc * Oc, Oc, 0LL, 0,
                Bh + (long long)l * Oc, 0, 1,
                hw, 0LL, 1, 0LL, (long long)Oc,
                Nc * Bc_, Oc, Fc);
        }
        // K6: fused GRU + sum over d + relu + highway
        {
            const int total = Nc * Bc_ * Oc;
            gru_highway_fuse<<<(total + 255) / 256, 256, 0, stream>>>(
                gi, gh, hw, xsrc, xdst);
        }
    }
}